// NGCF_22703197127156
// MI455X (gfx1250) — compile-verified
//
#include <hip/hip_runtime.h>
#include <hip/hip_bf16.h>

#define N_USERS  50000
#define N_ITEMS  50000
#define N_NODES  (N_USERS + N_ITEMS)
#define NNZ      1600000
#define EMB      64
#define N_LAYERS 3
#define REG_L2   1e-5f
#define BATCH    4096
#define TILES_M  (N_NODES / 16)   /* 6250 exact */

typedef __attribute__((ext_vector_type(2))) float v2f;
typedef __attribute__((ext_vector_type(8))) float v8f;

// ---------------------------------------------------------------------------
// Init: ego = concat(user_emb, item_emb); also write layer-0 slice of allE.
// ---------------------------------------------------------------------------
__global__ __launch_bounds__(256) void ngcf_init_kernel(
    const float* __restrict__ uemb, const float* __restrict__ iemb,
    float* __restrict__ ego, float* __restrict__ allE)
{
    size_t idx = (size_t)blockIdx.x * 256 + threadIdx.x;   // over N_NODES*64
    if (idx >= (size_t)N_NODES * EMB) return;
    size_t r = idx >> 6;
    int    c = (int)(idx & 63);
    float  v = (r < N_USERS) ? uemb[idx] : iemb[idx - (size_t)N_USERS * EMB];
    ego[idx] = v;
    allE[r * 256 + c] = v;
}

// ---------------------------------------------------------------------------
// SpMM scatter: side[row[e]] += val[e] * ego[col[e]].  One wave per edge,
// float2 per lane, hardware f32 global atomics (L2-resident gather).
// ---------------------------------------------------------------------------
__global__ __launch_bounds__(256) void ngcf_spmm_kernel(
    const int* __restrict__ arow, const int* __restrict__ acol,
    const float* __restrict__ aval, const float* __restrict__ x,
    float* __restrict__ y)
{
    int e  = blockIdx.x * 8 + (threadIdx.x >> 5);
    if (e >= NNZ) return;                         // whole-wave uniform
    int ln = threadIdx.x & 31;

    int   r = arow[e];
    int   c = acol[e];
    float v = aval[e];

    // prefetch next edge's source row (global_prefetch_b8)
    if (e + 8 < NNZ) {
        int cn = acol[e + 8];
        __builtin_prefetch(x + (size_t)cn * EMB + ln * 2, 0, 3);
    }

    v2f xv = *(const v2f*)(x + (size_t)c * EMB + ln * 2);
    float* yr = y + (size_t)r * EMB + ln * 2;
    unsafeAtomicAdd(yr,     v * xv.x);
    unsafeAtomicAdd(yr + 1, v * xv.y);
}

// ---------------------------------------------------------------------------
// Fused layer: sum = side@Wgc + bgc ; bi = (ego*(side-ego))@Wbi + bbi ;
// act = leaky_relu(sum+bi) ; write act (next ego, into side buffer) and
// act/||act|| into allE[:, layer_col:layer_col+64].
//
// One wave per 16-row tile. V_WMMA_F32_16X16X4_F32, K accumulated in 16
// steps of 4.  A frag (16x4 f32): lanes 0-15 hold K=0,1 ; lanes 16-31 K=2,3.
// B frag (4x16): lanes 0-15 hold K=0,1 ; lanes 16-31 K=2,3 (column = lane&15).
// W is pre-packed in LDS so a B fragment is a single aligned 8-byte ds load:
//   lds[kk*256 + n*4 + (kd&3)] = W[kd][n],  kk = kd>>2.
// ---------------------------------------------------------------------------
__global__ __launch_bounds__(256) void ngcf_layer_kernel(
    const float* __restrict__ side, const float* __restrict__ ego,
    const float* __restrict__ Wgc, const float* __restrict__ bgc,
    const float* __restrict__ Wbi, const float* __restrict__ bbi,
    float* __restrict__ act_out, float* __restrict__ allE, int layer_col)
{
    __shared__ float lgc[4096];
    __shared__ float lbi[4096];
    for (int t = threadIdx.x; t < 4096; t += 256) {
        int kd = t >> 6, n = t & 63;
        int idx = (kd >> 2) * 256 + n * 4 + (kd & 3);
        lgc[idx] = Wgc[t];
        lbi[idx] = Wbi[t];
    }
    __syncthreads();

    int tile = blockIdx.x * 8 + (threadIdx.x >> 5);
    if (tile >= TILES_M) return;                  // wave-uniform; EXEC all-1s below
    int ln   = threadIdx.x & 31;
    int half = ln >> 4;                           // 0: K=0,1 | 1: K=2,3
    int l15  = ln & 15;

    size_t rowbase = ((size_t)tile * 16 + l15) * EMB + half * 2;
    const float* sA = side + rowbase;
    const float* eA = ego  + rowbase;

    v8f accgc[4] = {{0,0,0,0,0,0,0,0},{0,0,0,0,0,0,0,0},
                    {0,0,0,0,0,0,0,0},{0,0,0,0,0,0,0,0}};
    v8f accbi[4] = {{0,0,0,0,0,0,0,0},{0,0,0,0,0,0,0,0},
                    {0,0,0,0,0,0,0,0},{0,0,0,0,0,0,0,0}};

    #pragma unroll 4
    for (int kk = 0; kk < 16; ++kk) {
        v2f as = *(const v2f*)(sA + kk * 4);
        v2f ae = *(const v2f*)(eA + kk * 4);
        v2f ab = ae * (as - ae);                  // ego * (side - ego)
        const float* bg = lgc + kk * 256 + half * 2;
        const float* bb = lbi + kk * 256 + half * 2;
        #pragma unroll
        for (int t = 0; t < 4; ++t) {
            v2f bgf = *(const v2f*)(bg + (t * 16 + l15) * 4);
            v2f bbf = *(const v2f*)(bb + (t * 16 + l15) * 4);
            accgc[t] = __builtin_amdgcn_wmma_f32_16x16x4_f32(
                false, as, false, bgf, (short)0, accgc[t], false, false);
            accbi[t] = __builtin_amdgcn_wmma_f32_16x16x4_f32(
                false, ab, false, bbf, (short)0, accbi[t], false, false);
        }
    }

    // Epilogue: bias + leaky_relu, then per-row L2 norm.
    // C/D layout: VGPR v -> row (tile*16 + v + half*8), column = t*16 + l15.
    float cb[4];
    #pragma unroll
    for (int t = 0; t < 4; ++t)
        cb[t] = bgc[t * 16 + l15] + bbi[t * 16 + l15];

    v8f act[4];
    float sq[8] = {0,0,0,0,0,0,0,0};
    #pragma unroll
    for (int t = 0; t < 4; ++t) {
        #pragma unroll
        for (int v = 0; v < 8; ++v) {
            float xv = accgc[t][v] + accbi[t][v] + cb[t];
            xv = (xv > 0.0f) ? xv : 0.01f * xv;   // leaky_relu(0.01)
            act[t][v] = xv;
            sq[v] += xv * xv;
        }
    }
    float scale[8];
    #pragma unroll
    for (int v = 0; v < 8; ++v) {
        float s = sq[v];                           // reduce over 16-lane half
        s += __shfl_xor(s, 1, 32);
        s += __shfl_xor(s, 2, 32);
        s += __shfl_xor(s, 4, 32);
        s += __shfl_xor(s, 8, 32);
        scale[v] = 1.0f / fmaxf(sqrtf(s), 1e-12f);
    }
    #pragma unroll
    for (int t = 0; t < 4; ++t) {
        int col = t * 16 + l15;
        #pragma unroll
        for (int v = 0; v < 8; ++v) {
            size_t m = (size_t)tile * 16 + v + half * 8;
            float  a = act[t][v];
            act_out[m * EMB + col]               = a;           // next ego
            allE[m * 256 + layer_col + col]      = a * scale[v]; // normalized
        }
    }
}

// ---------------------------------------------------------------------------
// BPR loss over sampled triples. One wave per sample, 8 floats/lane over the
// 256-wide concat rows; full-wave reduction; two global f32 atomic scalars.
// ---------------------------------------------------------------------------
__global__ __launch_bounds__(256) void ngcf_bpr_kernel(
    const float* __restrict__ allE, const int* __restrict__ u,
    const int* __restrict__ ii, const int* __restrict__ jj,
    float* __restrict__ acc)
{
    int b  = blockIdx.x * 8 + (threadIdx.x >> 5);
    if (b >= BATCH) return;
    int ln = threadIdx.x & 31;

    const float* ur = allE + (size_t)u[b] * 256 + ln * 8;
    const float* pr = allE + ((size_t)N_USERS + ii[b]) * 256 + ln * 8;
    const float* nr = allE + ((size_t)N_USERS + jj[b]) * 256 + ln * 8;

    float yui = 0.f, yuj = 0.f, l2 = 0.f;
    #pragma unroll
    for (int q = 0; q < 8; ++q) {
        float uv = ur[q], pv = pr[q], nv = nr[q];
        yui += uv * pv;
        yuj += uv * nv;
        l2  += uv * uv + pv * pv + nv * nv;
    }
    #pragma unroll
    for (int off = 1; off < 32; off <<= 1) {
        yui += __shfl_xor(yui, off, 32);
        yuj += __shfl_xor(yuj, off, 32);
        l2  += __shfl_xor(l2,  off, 32);
    }
    if (ln == 0) {
        float d  = yui - yuj;
        float ls = fminf(d, 0.0f) - log1pf(expf(-fabsf(d)));   // log_sigmoid(d)
        unsafeAtomicAdd(acc,     -ls);        // sum of -log_sigmoid
        unsafeAtomicAdd(acc + 1, 0.5f * l2);  // sum of squared norms / 2
    }
}

__global__ void ngcf_fin_kernel(const float* __restrict__ acc,
                                float* __restrict__ out)
{
    out[0] = acc[0] / (float)BATCH + REG_L2 * (acc[1] / (float)BATCH);
}

// ---------------------------------------------------------------------------
extern "C" void kernel_launch(void* const* d_in, const int* in_sizes, int n_in,
                              void* d_out, int out_size, void* d_ws, size_t ws_size,
                              hipStream_t stream)
{
    const float* user_emb = (const float*)d_in[0];
    const float* item_emb = (const float*)d_in[1];
    const float* W_gc     = (const float*)d_in[2];
    const float* b_gc     = (const float*)d_in[3];
    const float* W_bi     = (const float*)d_in[4];
    const float* b_bi     = (const float*)d_in[5];
    const float* adj_val  = (const float*)d_in[6];
    const int*   adj_row  = (const int*)d_in[7];
    const int*   adj_col  = (const int*)d_in[8];
    const int*   u        = (const int*)d_in[9];
    const int*   i_       = (const int*)d_in[10];
    const int*   j_       = (const int*)d_in[11];

    const size_t NODE64 = (size_t)N_NODES * EMB;      // 6.4M floats
    float* buf0 = (float*)d_ws;
    float* buf1 = buf0 + NODE64;
    float* allE = buf1 + NODE64;                      // N_NODES x 256
    float* acc  = allE + (size_t)N_NODES * 256;       // 2 floats

    // layer-0 embeddings
    {
        int blocks = (int)((NODE64 + 255) / 256);
        ngcf_init_kernel<<<blocks, 256, 0, stream>>>(user_emb, item_emb, buf0, allE);
    }

    float* cur   = buf0;   // ego input
    float* other = buf1;   // side / act output (ping-pong)
    for (int k = 0; k < N_LAYERS; ++k) {
        hipMemsetAsync(other, 0, NODE64 * sizeof(float), stream);
        ngcf_spmm_kernel<<<NNZ / 8, 256, 0, stream>>>(adj_row, adj_col, adj_val,
                                                      cur, other);
        ngcf_layer_kernel<<<(TILES_M + 7) / 8, 256, 0, stream>>>(
            other, cur,
            W_gc + (size_t)k * EMB * EMB, b_gc + (size_t)k * EMB,
            W_bi + (size_t)k * EMB * EMB, b_bi + (size_t)k * EMB,
            other, allE, 64 * (k + 1));
        float* t = cur; cur = other; other = t;
    }

    hipMemsetAsync(acc, 0, 2 * sizeof(float), stream);
    ngcf_bpr_kernel<<<BATCH / 8, 256, 0, stream>>>(allE, u, i_, j_, acc);
    ngcf_fin_kernel<<<1, 1, 0, stream>>>(acc, (float*)d_out);
}